// TemporalAttentionLayer_24318104830053
// MI455X (gfx1250) — compile-verified
//
#include <hip/hip_runtime.h>
#include <math.h>

typedef __attribute__((ext_vector_type(2))) float v2f;
typedef __attribute__((ext_vector_type(8))) float v8f;

#define ALPHA 0.2f
#define BB 4
#define KK 256
#define FF 128
#define EE 256          // 2*F
#define ITILE 4
#define ECHUNK 32

// ---------------------------------------------------------------------------
// Kernel 1: S[s][b*K+k, e] = sum_f X[b,k,f] * W[e, s*F + f]   (+ b_lin[e] if s==0)
// One 16x16 output tile per wave, V_WMMA_F32_16X16X4_F32, 32 k-steps.
// ---------------------------------------------------------------------------
__global__ __launch_bounds__(256)
void gat_gemm_s_kernel(const float* __restrict__ X,    // (1024,128)
                       const float* __restrict__ W,    // (256,256)
                       const float* __restrict__ blin, // (256)
                       float* __restrict__ S)          // (2,1024,256)
{
    const int wave = threadIdx.x >> 5;
    const int lane = threadIdx.x & 31;
    const int ln   = lane & 15;     // A: row M ; B: col N ; C/D: col N
    const int hi   = lane >> 4;     // K-pair selector / M+8 selector
    const int s    = blockIdx.y;    // 0 -> S1 (W[:, :F]),  1 -> S2 (W[:, F:])
    const int tile = blockIdx.x * 8 + wave;       // 0..1023
    const int row0 = (tile >> 4) * 16;            // M base (0..1008)
    const int col0 = (tile & 15) * 16;            // N base (0..240)

    const float* Arow = X + (size_t)(row0 + ln) * FF;
    const float* Brow = W + (size_t)(col0 + ln) * (2 * FF) + s * FF;

    v8f c = {};
    #pragma unroll 4
    for (int k0 = 0; k0 < FF; k0 += 4) {
        v2f a = *(const v2f*)(Arow + k0 + 2 * hi);   // K = k0+2hi, k0+2hi+1
        v2f b = *(const v2f*)(Brow + k0 + 2 * hi);
        c = __builtin_amdgcn_wmma_f32_16x16x4_f32(false, a, false, b,
                                                  (short)0, c, false, false);
    }

    const float add = (s == 0) ? blin[col0 + ln] : 0.0f;  // fold b_lin into S1
    float* D = S + (size_t)s * 1024 * EE;
    #pragma unroll
    for (int r = 0; r < 8; ++r)
        D[(size_t)(row0 + r + 8 * hi) * EE + col0 + ln] = c[r] + add;
}

// ---------------------------------------------------------------------------
// Kernel 2: the non-GEMM core.
//   e[b,i,j] = sum_e a[e]*lrelu(S1[b,i,e] + S2[b,j,e]) + bias[i,j]
//   attn[b,i,:] = softmax_j(e)
// Block = (b, 4 rows of i); thread j owns column j. S2[b] is streamed through
// LDS in 32-wide e-chunks (padded to 33 against bank conflicts) and reused
// across the 4 i-rows.
// ---------------------------------------------------------------------------
__global__ __launch_bounds__(256)
void gat_attn_kernel(const float* __restrict__ S,    // (2,1024,256), S1 has b_lin folded
                     const float* __restrict__ avec, // (256)
                     const float* __restrict__ bias, // (256,256)
                     float* __restrict__ attn)       // (4,256,256)
{
    const int b  = blockIdx.y;
    const int i0 = blockIdx.x * ITILE;
    const int j  = threadIdx.x;            // 0..255

    __shared__ float s1row[ITILE][EE];
    __shared__ float a_s[EE];
    __shared__ float tile[KK][ECHUNK + 1];
    __shared__ float red[KK];

    const float* S1 = S;
    const float* S2b = S + (size_t)1024 * EE + (size_t)b * KK * EE;

    a_s[j] = avec[j];
    #pragma unroll
    for (int ii = 0; ii < ITILE; ++ii)
        s1row[ii][j] = S1[(size_t)(b * KK + i0 + ii) * EE + j];
    __syncthreads();

    float acc[ITILE];
    #pragma unroll
    for (int ii = 0; ii < ITILE; ++ii) acc[ii] = 0.0f;

    for (int e0 = 0; e0 < EE; e0 += ECHUNK) {
        __syncthreads();                     // previous chunk fully consumed
        for (int idx = threadIdx.x; idx < KK * ECHUNK; idx += 256) {
            const int jj = idx >> 5;
            const int ee = idx & 31;
            tile[jj][ee] = S2b[(size_t)jj * EE + e0 + ee];   // coalesced
        }
        __syncthreads();
        #pragma unroll 8
        for (int ee = 0; ee < ECHUNK; ++ee) {
            const float t  = tile[j][ee];
            const float ae = a_s[e0 + ee];
            #pragma unroll
            for (int ii = 0; ii < ITILE; ++ii) {
                const float p = s1row[ii][e0 + ee] + t;
                const float h = (p >= 0.0f) ? p : ALPHA * p;
                acc[ii] = fmaf(ae, h, acc[ii]);
            }
        }
    }

    // per-row softmax over the 256 threads
    #pragma unroll
    for (int ii = 0; ii < ITILE; ++ii) {
        const float ev = acc[ii] + bias[(size_t)(i0 + ii) * KK + j];

        red[j] = ev;           __syncthreads();
        for (int off = 128; off > 0; off >>= 1) {
            if (j < off) red[j] = fmaxf(red[j], red[j + off]);
            __syncthreads();
        }
        const float mx = red[0]; __syncthreads();

        const float p = __expf(ev - mx);
        red[j] = p;            __syncthreads();
        for (int off = 128; off > 0; off >>= 1) {
            if (j < off) red[j] += red[j + off];
            __syncthreads();
        }
        const float inv = 1.0f / red[0];
        __syncthreads();       // red free before next ii

        attn[(size_t)(b * KK + i0 + ii) * KK + j] = p * inv;
    }
}

// ---------------------------------------------------------------------------
// Kernel 3: out[b] = sigmoid(attn[b] @ X[b]),  (256x256)@(256x128) per batch.
// WMMA f32 16x16x4, sigmoid fused in the epilogue.
// ---------------------------------------------------------------------------
__global__ __launch_bounds__(256)
void gat_out_kernel(const float* __restrict__ attn, // (4,256,256)
                    const float* __restrict__ X,    // (4,256,128)
                    float* __restrict__ out)        // (4,256,128)
{
    const int wave = threadIdx.x >> 5;
    const int lane = threadIdx.x & 31;
    const int ln   = lane & 15;
    const int hi   = lane >> 4;
    const int b    = blockIdx.y;
    const int tile = blockIdx.x * 8 + wave;    // 0..127
    const int row0 = (tile >> 3) * 16;         // 16 M-tiles
    const int col0 = (tile & 7) * 16;          // 8 N-tiles (F=128)

    const float* A  = attn + (size_t)b * KK * KK;
    const float* Bm = X    + (size_t)b * KK * FF;
    const float* Arow = A + (size_t)(row0 + ln) * KK;

    v8f c = {};
    #pragma unroll 4
    for (int k0 = 0; k0 < KK; k0 += 4) {
        v2f a = *(const v2f*)(Arow + k0 + 2 * hi);        // contiguous float2
        v2f bb;
        bb.x = Bm[(size_t)(k0 + 2 * hi) * FF + col0 + ln];      // K row k0+2hi
        bb.y = Bm[(size_t)(k0 + 2 * hi + 1) * FF + col0 + ln];  // K row +1
        c = __builtin_amdgcn_wmma_f32_16x16x4_f32(false, a, false, bb,
                                                  (short)0, c, false, false);
    }

    float* O = out + (size_t)b * KK * FF;
    #pragma unroll
    for (int r = 0; r < 8; ++r) {
        const float v = c[r];
        O[(size_t)(row0 + r + 8 * hi) * FF + col0 + ln] =
            1.0f / (1.0f + __expf(-v));
    }
}

// ---------------------------------------------------------------------------
extern "C" void kernel_launch(void* const* d_in, const int* in_sizes, int n_in,
                              void* d_out, int out_size, void* d_ws, size_t ws_size,
                              hipStream_t stream) {
    const float* x    = (const float*)d_in[0];  // (4,256,128)
    const float* W    = (const float*)d_in[1];  // (256,256)
    const float* blin = (const float*)d_in[2];  // (256,)
    const float* a    = (const float*)d_in[3];  // (256,)
    const float* bias = (const float*)d_in[4];  // (256,256)
    float* out = (float*)d_out;                 // (4,256,128)

    float* S     = (float*)d_ws;                        // 2 * 1024 * 256 f32 = 2 MB
    float* attnw = S + (size_t)2 * 1024 * EE;           // 4 * 256 * 256 f32 = 1 MB

    // S1/S2 GEMMs: 1024 tiles each, 8 waves/block
    gat_gemm_s_kernel<<<dim3(128, 2), 256, 0, stream>>>(x, W, blin, S);

    // LeakyReLU contraction + softmax: one block per (4 i-rows, batch)
    gat_attn_kernel<<<dim3(KK / ITILE, BB), 256, 0, stream>>>(S, a, bias, attnw);

    // attn @ x + sigmoid: 128 tiles/batch, 8 waves/block
    gat_out_kernel<<<dim3(16, BB), 256, 0, stream>>>(attnw, x, out);
}